// ShuntLayer_57604101374353
// MI455X (gfx1250) — compile-verified
//
#include <hip/hip_runtime.h>
#include <hip/hip_bf16.h>

// ---- problem constants (from reference) ----
#define BSZ   8
#define DD    512
#define NN    2048
#define EE    8
#define LL    64
#define NEDGE 524288
#define BN    (BSZ*NN)        // 16384 nodes
#define NET   (NEDGE+BN)      // 540672 edges incl self loops
#define C3EL  1536            // 3*E*L conv output channels
#define DM    49152           // N*3*E  (flattened (n,f) row length per (b,l))
#define KQ    16384           // N*E = attention feature dim per third

typedef __attribute__((ext_vector_type(2))) float v2f;
typedef __attribute__((ext_vector_type(8))) float v8f;

// V_WMMA_F32_16X16X4_F32 : D(16x16 f32) = A(16x4 f32) * B(4x16 f32) + C
__device__ __forceinline__ v8f wmma4(v2f a, v2f b, v8f c) {
  return __builtin_amdgcn_wmma_f32_16x16x4_f32(false, a, false, b, (short)0, c,
                                               false, false);
}

// order-preserving float<->uint map for atomic max on floats
__device__ __forceinline__ unsigned fenc(float x) {
  unsigned b = __float_as_uint(x);
  return (b & 0x80000000u) ? ~b : (b | 0x80000000u);
}
__device__ __forceinline__ float fdec(unsigned k) {
  unsigned b = (k & 0x80000000u) ? (k & 0x7FFFFFFFu) : ~k;
  return __uint_as_float(b);
}

// ---------------- stage 0: init accumulators ----------------
__global__ void k_init(unsigned* smax, float* ssum, float* h, float* S) {
  int i = blockIdx.x * blockDim.x + threadIdx.x;      // 131072 threads
  if (i < BN)        { smax[i] = 0x007FFFFFu; ssum[i] = 0.f; }  // enc(-inf)
  if (i < BN * EE)   h[i] = 0.f;
  if (i < BSZ*LL*LL) S[i] = 0.f;
}

// ---------------- stage 1: latent @ W_dense + b ----------------
__global__ void k_dense(const float* __restrict__ latent,
                        const float* __restrict__ W,
                        const float* __restrict__ bvec,
                        float* __restrict__ skel) {
  int bn = blockIdx.x * blockDim.x + threadIdx.x;
  if (bn >= BN) return;
  int b = bn >> 11, n = bn & (NN - 1);
  const float* lr = latent + b * DD;
  float acc = bvec[n];
  for (int k = 0; k < DD; ++k) acc += lr[k] * W[k * NN + n]; // lane-coalesced in n
  skel[bn] = acc;
}

// ---------------- stage 2: conv1(k=3) + GAT Q/K/V (8x8) ----------------
__global__ void k_nodeqkv(const float* __restrict__ skel,
                          const float* __restrict__ cw, const float* __restrict__ cb,
                          const float* __restrict__ Wq, const float* __restrict__ bq,
                          const float* __restrict__ Wk, const float* __restrict__ bk,
                          const float* __restrict__ Wv,
                          float* __restrict__ Q, float* __restrict__ Kf,
                          float* __restrict__ Vf) {
  int bn = blockIdx.x * blockDim.x + threadIdx.x;
  if (bn >= BN) return;
  int n = bn & (NN - 1);
  float sm = (n > 0)      ? skel[bn - 1] : 0.f;   // 'SAME' pad per batch row
  float s0 = skel[bn];
  float sp = (n < NN - 1) ? skel[bn + 1] : 0.f;
  float x[EE];
#pragma unroll
  for (int e = 0; e < EE; ++e)
    x[e] = sm * cw[e] + s0 * cw[8 + e] + sp * cw[16 + e] + cb[e];
#pragma unroll
  for (int e = 0; e < EE; ++e) {
    float q = bq[e], k = bk[e], v = 0.f;
#pragma unroll
    for (int f = 0; f < EE; ++f) {
      q += x[f] * Wq[f * 8 + e];
      k += x[f] * Wk[f * 8 + e];
      v += x[f] * Wv[f * 8 + e];
    }
    q = q > 0.f ? q : 0.2f * q;                    // leaky_relu 0.2
    k = k > 0.f ? k : 0.2f * k;
    Q[bn * 8 + e] = q; Kf[bn * 8 + e] = k; Vf[bn * 8 + e] = v;
  }
}

// ---------------- stage 3: per-edge score + segment max ----------------
__global__ void k_escore(const int* __restrict__ ei,
                         const float* __restrict__ Q, const float* __restrict__ Kf,
                         float* __restrict__ score, unsigned* __restrict__ smax) {
  int i = blockIdx.x * blockDim.x + threadIdx.x;
  if (i >= NET) return;
  int row = (i < NEDGE) ? ei[i]          : (i - NEDGE);
  int col = (i < NEDGE) ? ei[NEDGE + i]  : (i - NEDGE);
  float s = 0.f;
#pragma unroll
  for (int e = 0; e < EE; ++e) s += Q[row * 8 + e] * Kf[col * 8 + e];
  score[i] = s;
  atomicMax(&smax[row], fenc(s));
}

// ---------------- stage 4: exp + segment sum ----------------
__global__ void k_eexp(const int* __restrict__ ei, const float* __restrict__ score,
                       const unsigned* __restrict__ smax,
                       float* __restrict__ ex, float* __restrict__ ssum) {
  int i = blockIdx.x * blockDim.x + threadIdx.x;
  if (i >= NET) return;
  int row = (i < NEDGE) ? ei[i] : (i - NEDGE);
  float ev = __expf(score[i] - fdec(smax[row]));
  ex[i] = ev;
  atomicAdd(&ssum[row], ev);
}

// ---------------- stage 5: alpha * V[col] scatter-add ----------------
__global__ void k_eagg(const int* __restrict__ ei, const float* __restrict__ ex,
                       const float* __restrict__ ssum, const float* __restrict__ Vf,
                       float* __restrict__ h) {
  int idx = blockIdx.x * blockDim.x + threadIdx.x;   // NET*8 threads
  int e = idx & 7, i = idx >> 3;
  if (i >= NET) return;
  int row = (i < NEDGE) ? ei[i]         : (i - NEDGE);
  int col = (i < NEDGE) ? ei[NEDGE + i] : (i - NEDGE);
  float alpha = ex[i] / ssum[row];
  atomicAdd(&h[row * 8 + e], alpha * Vf[col * 8 + e]);
}

// ---------------- stage 6: qkv conv as WMMA GEMM [2048x24]@[24x1536] ----------
// writes directly into D3[b][l][n*24+g]  (l=c/24, g=c%24)
__global__ void k_qkvgemm(const float* __restrict__ h, const float* __restrict__ gbias,
                          const float* __restrict__ qw, const float* __restrict__ qb,
                          float* __restrict__ D3) {
  int w    = (blockIdx.x * blockDim.x + threadIdx.x) >> 5;  // 98304 waves
  int lane = threadIdx.x & 31;
  int r    = lane & 15, half = lane >> 4;
  int nt   = w % 96;
  int rest = w / 96;
  int mt   = rest & 127;
  int b    = rest >> 7;
  int c = nt * 16 + r;          // B col / output channel for this lane
  int n = mt * 16 + r;          // A row / node for this lane
  v8f acc = {};
#pragma unroll
  for (int kk = 0; kk < 6; ++kk) {           // K=24, 6 steps of 4
    v2f a, bf;
#pragma unroll
    for (int v = 0; v < 2; ++v) {
      int f = kk * 4 + half * 2 + v;         // im2col column: tap t, channel e
      int t = f >> 3, e = f & 7;
      int src = n + t - 1;                   // 'SAME' conv window
      a[v] = (src >= 0 && src < NN) ? (h[((b << 11) + src) * 8 + e] + gbias[e])
                                    : 0.f;
      bf[v] = qw[t * 12288 + e * 1536 + c];  // qkv_w[t][e][c]
    }
    acc = wmma4(a, bf, acc);
  }
  float bias = qb[c];
  int l = c / 24, g = c - l * 24;
#pragma unroll
  for (int rr = 0; rr < 8; ++rr) {
    int row = mt * 16 + rr + half * 8;       // C layout: M = vgpr + half*8
    D3[((size_t)(b * LL + l)) * DM + (size_t)row * 24 + g] = acc[rr] + bias;
  }
}

// ---------------- stage 7: S = q @ k^T  (K=16384 split over 64 chunks) --------
__global__ void k_scores(const float* __restrict__ D3, float* __restrict__ S) {
  int w    = (blockIdx.x * blockDim.x + threadIdx.x) >> 5;  // 8192 waves
  int lane = threadIdx.x & 31;
  int r    = lane & 15, half = lane >> 4;
  int kc = w & 63, nt = (w >> 6) & 3, mt = (w >> 8) & 3, b = w >> 10;
  const float* Qb = D3 + (size_t)b * LL * DM;
  int i = mt * 16 + r;          // q row
  int j = nt * 16 + r;          // k row (B col)
  v8f acc = {};
  int k0base = kc * 256;
  for (int ks = 0; ks < 64; ++ks) {
    int k0 = k0base + ks * 4;
    v2f a, bf;
#pragma unroll
    for (int v = 0; v < 2; ++v) {
      int kidx = k0 + half * 2 + v;
      a[v]  = Qb[(size_t)i * DM + kidx];              // q third: offset 0
      bf[v] = Qb[(size_t)j * DM + 32768 + kidx];      // k third: offset 2*KQ
    }
    acc = wmma4(a, bf, acc);
  }
#pragma unroll
  for (int rr = 0; rr < 8; ++rr) {
    int row = mt * 16 + rr + half * 8;
    atomicAdd(&S[(size_t)(b * LL + row) * LL + nt * 16 + r], acc[rr]);
  }
}

// ---------------- stage 8: row softmax on S [512 rows x 64] ----------------
__global__ void k_softmax(float* __restrict__ S) {
  int row  = (blockIdx.x * blockDim.x + threadIdx.x) >> 5;  // 512 waves
  int lane = threadIdx.x & 31;
  float v0 = S[row * 64 + lane];
  float v1 = S[row * 64 + 32 + lane];
  float m = fmaxf(v0, v1);
  for (int o = 16; o > 0; o >>= 1) m = fmaxf(m, __shfl_xor(m, o, 32));
  float e0 = __expf(v0 - m), e1 = __expf(v1 - m);
  float s = e0 + e1;
  for (int o = 16; o > 0; o >>= 1) s += __shfl_xor(s, o, 32);
  S[row * 64 + lane]      = e0 / s;
  S[row * 64 + 32 + lane] = e1 / s;
}

// ---------------- stage 9: out = A[64x64] @ V[64x16384] ----------------
__global__ void k_seq(const float* __restrict__ S, const float* __restrict__ D3,
                      float* __restrict__ out) {
  int w    = (blockIdx.x * blockDim.x + threadIdx.x) >> 5;  // 32768 waves
  int lane = threadIdx.x & 31;
  int r    = lane & 15, half = lane >> 4;
  int nt = w & 1023, mt = (w >> 10) & 3, b = w >> 12;
  const float* Vb = D3 + (size_t)b * LL * DM + KQ;          // v third: offset KQ
  int j = nt * 16 + r;
  v8f acc = {};
#pragma unroll
  for (int kk = 0; kk < 16; ++kk) {          // K = 64
    v2f a, bf;
#pragma unroll
    for (int v = 0; v < 2; ++v) {
      int k = kk * 4 + half * 2 + v;
      a[v]  = S[(size_t)(b * LL + mt * 16 + r) * LL + k];
      bf[v] = Vb[(size_t)k * DM + j];        // contiguous in j across lanes
    }
    acc = wmma4(a, bf, acc);
  }
#pragma unroll
  for (int rr = 0; rr < 8; ++rr) {
    int l = mt * 16 + rr + half * 8;
    out[((size_t)(b * LL + l)) * KQ + j] = acc[rr];  // [b,l,n,e] == [b,l,n*8+e]
  }
}

extern "C" void kernel_launch(void* const* d_in, const int* in_sizes, int n_in,
                              void* d_out, int out_size, void* d_ws, size_t ws_size,
                              hipStream_t stream) {
  const float* latent   = (const float*)d_in[0];
  const int*   eidx     = (const int*)  d_in[1];
  const float* W_dense  = (const float*)d_in[2];
  const float* b_dense  = (const float*)d_in[3];
  const float* conv1_w  = (const float*)d_in[4];
  const float* conv1_b  = (const float*)d_in[5];
  const float* gat_Wq   = (const float*)d_in[6];
  const float* gat_bq   = (const float*)d_in[7];
  const float* gat_Wk   = (const float*)d_in[8];
  const float* gat_bk   = (const float*)d_in[9];
  const float* gat_Wv   = (const float*)d_in[10];
  const float* gat_bias = (const float*)d_in[11];
  const float* qkv_w    = (const float*)d_in[12];
  const float* qkv_b    = (const float*)d_in[13];
  float* out = (float*)d_out;

  // workspace carve-up (floats)
  float*    wsf   = (float*)d_ws;
  float*    Q     = wsf;                 // 131072
  float*    Kf    = Q    + 131072;       // 131072
  float*    Vf    = Kf   + 131072;       // 131072
  float*    h     = Vf   + 131072;       // 131072
  float*    skel  = h    + 131072;       // 16384
  unsigned* smax  = (unsigned*)(skel + 16384);   // 16384
  float*    ssum  = (float*)(smax + 16384);      // 16384
  float*    score = ssum + 16384;        // 540672
  float*    ex    = score + 540672;      // 540672
  float*    S     = ex   + 540672;       // 32768
  float*    D3    = S    + 32768;        // 25165824 (~100 MB)

  dim3 blk(256);
  k_init   <<<  512, blk, 0, stream>>>(smax, ssum, h, S);
  k_dense  <<<   64, blk, 0, stream>>>(latent, W_dense, b_dense, skel);
  k_nodeqkv<<<   64, blk, 0, stream>>>(skel, conv1_w, conv1_b, gat_Wq, gat_bq,
                                       gat_Wk, gat_bk, gat_Wv, Q, Kf, Vf);
  k_escore <<< 2112, blk, 0, stream>>>(eidx, Q, Kf, score, smax);
  k_eexp   <<< 2112, blk, 0, stream>>>(eidx, score, smax, ex, ssum);
  k_eagg   <<<16896, blk, 0, stream>>>(eidx, ex, ssum, Vf, h);
  k_qkvgemm<<<12288, blk, 0, stream>>>(h, gat_bias, qkv_w, qkv_b, D3);
  k_scores <<< 1024, blk, 0, stream>>>(D3, S);
  k_softmax<<<   64, blk, 0, stream>>>(S);
  k_seq    <<< 4096, blk, 0, stream>>>(S, D3, out);
}